// MLPPredictor_11269994185514
// MI455X (gfx1250) — compile-verified
//
#include <hip/hip_runtime.h>
#include <hip/hip_bf16.h>

typedef __attribute__((ext_vector_type(2))) float v2f;
typedef __attribute__((ext_vector_type(8))) float v8f;

#define D_FEAT   128
#define IN_DIM   256
#define HIDDEN   64
#define EDGES_PER_BLOCK 128
#define SX 260   // padded row stride (floats) for X tile: 1040B, 16B aligned, bank-conflict-free
#define SW 260   // padded row stride (floats) for W1^T

__global__ __launch_bounds__(256) void mlp_edge_wmma_kernel(
    const float* __restrict__ emb,        // [nNodes][128]
    const long long* __restrict__ ts,     // [nEdges][2]
    const float* __restrict__ w1,         // [256][64] row-major
    const float* __restrict__ b1,         // [64]
    const float* __restrict__ w2,         // [64]
    float* __restrict__ out,              // [nEdges]
    int nEdges, int nNodes)
{
    extern __shared__ float smem[];
    float* Xs = smem;                         // [128][SX]
    float* Ws = smem + EDGES_PER_BLOCK * SX;  // [64][SW]  (W1 transposed: Ws[n][k])

    const int tid = threadIdx.x;
    const int e0  = blockIdx.x * EDGES_PER_BLOCK;

    // ---------------- Stage W1^T into LDS ----------------
    // thread t: column n = t&63, k-block = (t>>6)*64
    {
        const int n  = tid & 63;
        const int kb = (tid >> 6) << 6;
        #pragma unroll 8
        for (int kk = 0; kk < 64; ++kk) {
            Ws[n * SW + kb + kk] = w1[(kb + kk) * HIDDEN + n];
        }
    }

    // ---------------- Gather edge features into LDS ----------------
    // thread t: local edge = t>>1, half = t&1 (0 -> src feats, 1 -> dst feats)
    {
        const int el   = tid >> 1;
        const int half = tid & 1;
        int e = e0 + el;
        if (e >= nEdges) e = nEdges - 1;          // clamp (stores are guarded later)
        long long node = ts[2LL * e + half];
        if (node < 0) node = 0;
        if (node >= nNodes) node = nNodes - 1;
        const float4* srcv = (const float4*)(emb + node * (long long)D_FEAT);
        float4* dstv = (float4*)(Xs + el * SX + half * D_FEAT);
        #pragma unroll
        for (int i = 0; i < D_FEAT / 4; ++i) dstv[i] = srcv[i];
    }

    __syncthreads();

    // ---------------- WMMA: 16 edges x 64 hidden per wave ----------------
    const int lane = tid & 31;
    const int wv   = tid >> 5;        // wave 0..7 -> edge rows wv*16..wv*16+15
    const int hl   = lane & 15;
    const int hi   = lane >> 4;       // 0: K{0,1}, 1: K{2,3} per ISA 16x4 f32 layout

    const float* Arow = Xs + (wv * 16 + hl) * SX + hi * 2;     // A: row = M, K striped
    const float* Bc0  = Ws + ( 0 + hl) * SW + hi * 2;          // B: lane = column N
    const float* Bc1  = Ws + (16 + hl) * SW + hi * 2;
    const float* Bc2  = Ws + (32 + hl) * SW + hi * 2;
    const float* Bc3  = Ws + (48 + hl) * SW + hi * 2;

    v8f acc0 = {}, acc1 = {}, acc2 = {}, acc3 = {};

    #pragma unroll 8
    for (int k = 0; k < IN_DIM; k += 4) {
        v2f a  = *(const v2f*)(Arow + k);
        v2f b0 = *(const v2f*)(Bc0 + k);
        v2f b1v = *(const v2f*)(Bc1 + k);
        v2f b2 = *(const v2f*)(Bc2 + k);
        v2f b3 = *(const v2f*)(Bc3 + k);
        acc0 = __builtin_amdgcn_wmma_f32_16x16x4_f32(false, a, false, b0,  (short)0, acc0, false, false);
        acc1 = __builtin_amdgcn_wmma_f32_16x16x4_f32(false, a, false, b1v, (short)0, acc1, false, false);
        acc2 = __builtin_amdgcn_wmma_f32_16x16x4_f32(false, a, false, b2,  (short)0, acc2, false, false);
        acc3 = __builtin_amdgcn_wmma_f32_16x16x4_f32(false, a, false, b3,  (short)0, acc3, false, false);
    }

    // ---------------- Epilogue: +b1, ReLU, dot w2, reduce over N ----------------
    // C/D layout: VGPR j, lanes 0-15 -> (M=j, N=lane); lanes 16-31 -> (M=8+j, N=lane-16)
    float bb0 = b1[ 0 + hl], bb1 = b1[16 + hl], bb2 = b1[32 + hl], bb3 = b1[48 + hl];
    float ww0 = w2[ 0 + hl], ww1 = w2[16 + hl], ww2 = w2[32 + hl], ww3 = w2[48 + hl];

    float res[8];
    #pragma unroll
    for (int j = 0; j < 8; ++j) {
        float s = fmaxf(acc0[j] + bb0, 0.0f) * ww0
                + fmaxf(acc1[j] + bb1, 0.0f) * ww1
                + fmaxf(acc2[j] + bb2, 0.0f) * ww2
                + fmaxf(acc3[j] + bb3, 0.0f) * ww3;
        // reduce across the 16 lanes of this half (xor masks < 16 stay in-half)
        s += __shfl_xor(s, 1, 32);
        s += __shfl_xor(s, 2, 32);
        s += __shfl_xor(s, 4, 32);
        s += __shfl_xor(s, 8, 32);
        res[j] = s;
    }

    if (hl == 0) {
        #pragma unroll
        for (int j = 0; j < 8; ++j) {
            int e = e0 + wv * 16 + hi * 8 + j;   // M = hi*8 + j
            if (e < nEdges) out[e] = res[j];
        }
    }
}

extern "C" void kernel_launch(void* const* d_in, const int* in_sizes, int n_in,
                              void* d_out, int out_size, void* d_ws, size_t ws_size,
                              hipStream_t stream) {
    const float*     emb = (const float*)d_in[0];
    const long long* ts  = (const long long*)d_in[1];
    const float*     w1  = (const float*)d_in[2];
    const float*     b1  = (const float*)d_in[3];
    const float*     w2  = (const float*)d_in[4];
    float* out = (float*)d_out;

    const int nNodes = in_sizes[0] / D_FEAT;
    const int nEdges = in_sizes[1] / 2;

    const int blocks = (nEdges + EDGES_PER_BLOCK - 1) / EDGES_PER_BLOCK;
    const size_t smemBytes = (size_t)(EDGES_PER_BLOCK * SX + HIDDEN * SW) * sizeof(float);

    mlp_edge_wmma_kernel<<<blocks, 256, smemBytes, stream>>>(
        emb, ts, w1, b1, w2, out, nEdges, nNodes);
}